// AttentionalPropagation_54795192762709
// MI455X (gfx1250) — compile-verified
//
#include <hip/hip_runtime.h>
#include <hip/hip_bf16.h>

typedef float v2f __attribute__((ext_vector_type(2)));
typedef float v8f __attribute__((ext_vector_type(8)));

#define DD   128
#define NQ_  256
#define NK_  2048
#define NMNK (NQ_ * NK_)   // 524288 = stride between dist channels
#define SSTR 18            // slab row stride (16 k-values + 2 pad) -> conflict-free b64 reads

#if __has_builtin(__builtin_amdgcn_global_load_async_to_lds_b32)
#define ASYNC_LDS 1
#else
#define ASYNC_LDS 0
#endif

__device__ __forceinline__ void wait_async_le8() {
#if ASYNC_LDS
#if __has_builtin(__builtin_amdgcn_s_wait_asynccnt)
  __builtin_amdgcn_s_wait_asynccnt(8);
#else
  asm volatile("s_wait_asynccnt 0x8" ::: "memory");
#endif
#endif
}
__device__ __forceinline__ void wait_async_le0() {
#if ASYNC_LDS
#if __has_builtin(__builtin_amdgcn_s_wait_asynccnt)
  __builtin_amdgcn_s_wait_asynccnt(0);
#else
  asm volatile("s_wait_asynccnt 0x0" ::: "memory");
#endif
#endif
}

// ---------------------------------------------------------------- init
__global__ void k0_init_min(unsigned int* minu) { *minu = 0xFFFFFFFFu; }

// ---------------------------------------------------------------- K1: qhb_t[n][o] = (wq @ x)[o,n] + b1[o]
__global__ __launch_bounds__(256) void k1_q(const float* __restrict__ x,
                                            const float* __restrict__ w1,
                                            const float* __restrict__ b1,
                                            float* __restrict__ qhb) {
  int idx = blockIdx.x * 256 + threadIdx.x;   // 128*256 outputs
  int o = idx >> 8, n = idx & 255;
  const float* wr = w1 + o * 384;             // wq = w1[:, :128]
  float acc = 0.f;
  for (int c = 0; c < DD; ++c) acc = fmaf(wr[c], x[c * NQ_ + n], acc);
  qhb[n * DD + o] = acc + b1[o];              // transposed for fast per-n staging
}

// ---------------------------------------------------------------- K1: kh[o][m] = (wk @ source)[o,m]
__global__ __launch_bounds__(256) void k1_k(const float* __restrict__ source,
                                            const float* __restrict__ w1,
                                            float* __restrict__ kh) {
  int idx = blockIdx.x * 256 + threadIdx.x;   // 128*2048 outputs
  int o = idx >> 11, m = idx & 2047;
  const float* wr = w1 + o * 384 + 128;       // wk = w1[:, 128:256]
  float acc = 0.f;
  for (int c = 0; c < DD; ++c) acc = fmaf(wr[c], source[c * NK_ + m], acc);
  kh[o * NK_ + m] = acc;
}

// ---------------------------------------------------------------- K2: fused dh-GEMM + relu + w2 reduction -> raw scores + global min
// grid = (NK/128, NQ/16), block = 256 (8 waves). Wave w owns output channels [16w, 16w+16).
// dist is staged in a double-buffered transposed LDS slab via async global->LDS DMA.
__global__ __launch_bounds__(256) void k2_scores(const float* __restrict__ dist,
                                                 const float* __restrict__ w1,
                                                 const float* __restrict__ w2,
                                                 const float* __restrict__ b2p,
                                                 const float* __restrict__ qhb,
                                                 const float* __restrict__ kh,
                                                 float* __restrict__ raw,
                                                 unsigned int* __restrict__ minu) {
  __shared__ float qb_s[128];
  __shared__ float w2_s[128];
  __shared__ float sbuf[128];
  __shared__ float dslab[2][128 * SSTR];      // [buf][m*SSTR + k], 9216B each

  const int tid  = threadIdx.x;
  const int wave = tid >> 5, lane = tid & 31;
  const int mrow = lane & 15, hi = lane >> 4; // WMMA f32 lane geometry
  const int koff = hi * 2, chi = hi * 8;
  const int m0b  = blockIdx.x * 128;
  const int n0   = blockIdx.y * 16;

  if (tid < 128) w2_s[tid] = w2[tid];

  // A fragments: this wave's 16 rows of wd (= w1[:, 256:384]), resident in VGPRs for whole block.
  // A 16x4 layout: lane m = l&15; VGPR0 = K=k0+koff, VGPR1 = K=k0+koff+1.
  v2f afr[32];
  {
    const float* wrow = w1 + (wave * 16 + mrow) * 384 + 256;
#pragma unroll
    for (int i = 0; i < 32; ++i) {
      afr[i].x = wrow[4 * i + koff];
      afr[i].y = wrow[4 * i + koff + 1];
    }
  }
  const float b2v = b2p[0];
  float thrmin = 3.0e38f;

  // staging thread geometry: element i = tid + j*256 -> k-row r = i>>7, column m = i&127
  const int st_r = tid >> 7;       // base k-row for j=0 (0 or 1)
  const int st_m = tid & 127;

  for (int nn = 0; nn < 16; ++nn) {
    const int n = n0 + nn;
    __syncthreads();                               // protect prev-iter LDS reads
    if (tid < 128) { qb_s[tid] = qhb[n * DD + tid]; sbuf[tid] = 0.f; }

    v8f acc[8];
    const v8f z = {0.f, 0.f, 0.f, 0.f, 0.f, 0.f, 0.f, 0.f};
#pragma unroll
    for (int mt = 0; mt < 8; ++mt) acc[mt] = z;

    const float* distn = dist + (unsigned)(n * NK_ + m0b);   // uniform base (scalar)

    // stage one 16x128 chunk (transposed) into buffer `b`; 32-bit lane offsets -> SADDR+voffset form
    auto stage = [&](int kc, int b) {
#pragma unroll
      for (int j = 0; j < 8; ++j) {
        const int r = st_r + j * 2;                // k-row 0..15
        const unsigned gofs = (unsigned)(kc * 16 + r) * (unsigned)NMNK + (unsigned)st_m;
        const float* g = distn + gofs;
        float* l = &dslab[b][st_m * SSTR + r];
#if ASYNC_LDS
        __builtin_amdgcn_global_load_async_to_lds_b32(
            (__attribute__((address_space(1))) int*)(void*)g,
            (__attribute__((address_space(3))) int*)(void*)l, 0, 0);
#else
        *l = *g;
#endif
      }
    };

    stage(0, 0);                                   // prologue
#pragma unroll
    for (int kc = 0; kc < 8; ++kc) {               // K chunks of 16 channels
      if (kc < 7) { stage(kc + 1, (kc + 1) & 1); wait_async_le8(); }
      else        { wait_async_le0(); }
      __syncthreads();                             // chunk kc visible to all waves
      const float* bufm = &dslab[kc & 1][0];
#pragma unroll
      for (int k4 = 0; k4 < 4; ++k4) {             // 4 WMMA K-steps (K=4 each)
        const int k0l = k4 * 4;
        const v2f a = afr[kc * 4 + k4];
#pragma unroll
        for (int mt = 0; mt < 8; ++mt) {           // 8 m-tiles of 16 cols
          const v2f b = *(const v2f*)&bufm[(mt * 16 + mrow) * SSTR + k0l + koff];
          acc[mt] = __builtin_amdgcn_wmma_f32_16x16x4_f32(
              false, a, false, b, (short)0, acc[mt], false, false);
        }
      }
      __syncthreads();                             // WAR: buffer re-staged 2 chunks later
    }

    // Epilogue: h = relu(dh + qh[c,n]+b1[c] + kh[c,m]); partial = sum_c w2[c]*h
#pragma unroll
    for (int mt = 0; mt < 8; ++mt) {
      const int m = m0b + mt * 16 + mrow;
      float p = 0.f;
#pragma unroll
      for (int v = 0; v < 8; ++v) {                // C/D layout: row M = v + 8*hi
        const int c = wave * 16 + chi + v;
        float h = acc[mt][v] + qb_s[c] + kh[(unsigned)(c * NK_ + m)];
        h = fmaxf(h, 0.f);
        p = fmaf(w2_s[c], h, p);
      }
      p += __shfl_xor(p, 16);                      // fold lane halves (wave32)
      if (lane < 16) atomicAdd(&sbuf[mt * 16 + mrow], p);  // fold 8 c-block waves
    }
    __syncthreads();
    if (tid < 128) {
      const float sc = sbuf[tid] + b2v;
      raw[(size_t)n * NK_ + m0b + tid] = sc;
      thrmin = fminf(thrmin, sc);
    }
  }

  // block min -> one global atomic (ordered-uint mapping)
  __syncthreads();
  if (tid < 128) sbuf[tid] = thrmin;
  __syncthreads();
  if (tid == 0) {
    float mn = sbuf[0];
    for (int i = 1; i < 128; ++i) mn = fminf(mn, sbuf[i]);
    unsigned int b = __float_as_uint(mn);
    unsigned int u = (b & 0x80000000u) ? ~b : (b | 0x80000000u);
    atomicMin(minu, u);
  }
}

// ---------------------------------------------------------------- K3: mask + softmax (writes masked scores in place, prob to ws)
__global__ __launch_bounds__(256) void k3_softmax(float* __restrict__ scores,
                                                  const unsigned char* __restrict__ mask,
                                                  const unsigned int* __restrict__ minu,
                                                  float* __restrict__ prob) {
  __shared__ float red[256];
  const int n = blockIdx.x, tid = threadIdx.x;
  const unsigned int u = *minu;
  const unsigned int bb = (u & 0x80000000u) ? (u ^ 0x80000000u) : ~u;
  const float neg = __uint_as_float(bb) - 20.f;

  float vals[8];
  float mx = -3.0e38f;
#pragma unroll
  for (int j = 0; j < 8; ++j) {
    const size_t idx = (size_t)n * NK_ + tid + j * 256;
    float v = scores[idx];
    if (!mask[idx]) v += neg;
    scores[idx] = v;                    // returned (masked) scores
    vals[j] = v;
    mx = fmaxf(mx, v);
  }
  red[tid] = mx; __syncthreads();
  for (int off = 128; off > 0; off >>= 1) {
    if (tid < off) red[tid] = fmaxf(red[tid], red[tid + off]);
    __syncthreads();
  }
  mx = red[0]; __syncthreads();
  float s = 0.f;
#pragma unroll
  for (int j = 0; j < 8; ++j) { vals[j] = __expf(vals[j] - mx); s += vals[j]; }
  red[tid] = s; __syncthreads();
  for (int off = 128; off > 0; off >>= 1) {
    if (tid < off) red[tid] += red[tid + off];
    __syncthreads();
  }
  const float inv = 1.f / red[0];
#pragma unroll
  for (int j = 0; j < 8; ++j)
    prob[(size_t)n * NK_ + tid + j * 256] = vals[j] * inv;
}

// ---------------------------------------------------------------- K4: message[d][n] = sum_m prob[n,m]*source[d,m]
__global__ __launch_bounds__(256) void k4_message(const float* __restrict__ prob,
                                                  const float* __restrict__ source,
                                                  float* __restrict__ msg) {
  __shared__ float As[16][17], Bs[16][17];
  const int tx = threadIdx.x & 15, ty = threadIdx.x >> 4;
  const int d0 = blockIdx.x * 16, n0 = blockIdx.y * 16;
  float acc = 0.f;
  for (int m0 = 0; m0 < NK_; m0 += 16) {
    As[ty][tx] = source[(size_t)(d0 + ty) * NK_ + m0 + tx];
    Bs[ty][tx] = prob[(size_t)(n0 + ty) * NK_ + m0 + tx];
    __syncthreads();
#pragma unroll
    for (int k = 0; k < 16; ++k) acc = fmaf(As[ty][k], Bs[tx][k], acc);
    __syncthreads();
  }
  msg[(d0 + ty) * NQ_ + n0 + tx] = acc;
}

// ---------------------------------------------------------------- K5a: hid = relu(w3 @ [x; message] + b3)
__global__ __launch_bounds__(256) void k5_hidden(const float* __restrict__ x,
                                                 const float* __restrict__ msg,
                                                 const float* __restrict__ w3,
                                                 const float* __restrict__ b3,
                                                 float* __restrict__ hid) {
  __shared__ float As[16][17], Bs[16][17];
  const int tx = threadIdx.x & 15, ty = threadIdx.x >> 4;
  const int o0 = blockIdx.x * 16, n0 = blockIdx.y * 16;
  float acc = 0.f;
  for (int c0 = 0; c0 < 256; c0 += 16) {
    As[ty][tx] = w3[(o0 + ty) * 256 + c0 + tx];
    const int c = c0 + ty, n = n0 + tx;
    Bs[ty][tx] = (c < 128) ? x[c * NQ_ + n] : msg[(c - 128) * NQ_ + n];
    __syncthreads();
#pragma unroll
    for (int k = 0; k < 16; ++k) acc = fmaf(As[ty][k], Bs[k][tx], acc);
    __syncthreads();
  }
  hid[(o0 + ty) * NQ_ + n0 + tx] = fmaxf(acc + b3[o0 + ty], 0.f);
}

// ---------------------------------------------------------------- K5b: out = w4 @ hid + b4
__global__ __launch_bounds__(256) void k5_out(const float* __restrict__ hid,
                                              const float* __restrict__ w4,
                                              const float* __restrict__ b4,
                                              float* __restrict__ out) {
  __shared__ float As[16][17], Bs[16][17];
  const int tx = threadIdx.x & 15, ty = threadIdx.x >> 4;
  const int o0 = blockIdx.x * 16, n0 = blockIdx.y * 16;
  float acc = 0.f;
  for (int c0 = 0; c0 < 256; c0 += 16) {
    As[ty][tx] = w4[(o0 + ty) * 256 + c0 + tx];
    Bs[ty][tx] = hid[(c0 + ty) * NQ_ + n0 + tx];
    __syncthreads();
#pragma unroll
    for (int k = 0; k < 16; ++k) acc = fmaf(As[ty][k], Bs[k][tx], acc);
    __syncthreads();
  }
  out[(o0 + ty) * NQ_ + n0 + tx] = acc + b4[o0 + ty];
}

// ---------------------------------------------------------------- launch
extern "C" void kernel_launch(void* const* d_in, const int* in_sizes, int n_in,
                              void* d_out, int out_size, void* d_ws, size_t ws_size,
                              hipStream_t stream) {
  const float*         x      = (const float*)d_in[0];
  const float*         source = (const float*)d_in[1];
  const float*         dist   = (const float*)d_in[2];
  const unsigned char* mask   = (const unsigned char*)d_in[3];
  const float*         w1     = (const float*)d_in[4];
  const float*         b1     = (const float*)d_in[5];
  const float*         w2     = (const float*)d_in[6];
  const float*         b2     = (const float*)d_in[7];
  const float*         w3     = (const float*)d_in[8];
  const float*         b3     = (const float*)d_in[9];
  const float*         w4     = (const float*)d_in[10];
  const float*         b4     = (const float*)d_in[11];

  float* out    = (float*)d_out;            // out[128*256]
  float* scores = out + 128 * 256;          // scores[256*2048] (raw, then masked in place)

  float* ws   = (float*)d_ws;
  float* qhb  = ws;                         //  32768 floats  (qh+b1, transposed [n][o])
  float* kh   = ws + 32768;                 // 262144 floats
  float* prob = ws + 294912;                // 524288 floats
  float* msg  = ws + 819200;                //  32768 floats
  float* hid  = ws + 851968;                //  65536 floats
  unsigned int* minu = (unsigned int*)(ws + 917504);

  k0_init_min<<<1, 1, 0, stream>>>(minu);
  k1_q<<<128, 256, 0, stream>>>(x, w1, b1, qhb);
  k1_k<<<1024, 256, 0, stream>>>(source, w1, kh);
  k2_scores<<<dim3(16, 16), 256, 0, stream>>>(dist, w1, w2, b2, qhb, kh, scores, minu);
  k3_softmax<<<256, 256, 0, stream>>>(scores, mask, minu, prob);
  k4_message<<<dim3(8, 16), 256, 0, stream>>>(prob, source, msg);
  k5_hidden<<<dim3(16, 16), 256, 0, stream>>>(x, msg, w3, b3, hid);
  k5_out<<<dim3(8, 16), 256, 0, stream>>>(hid, w4, b4, out);
}